// GMLVQ_26800595927194
// MI455X (gfx1250) — compile-verified
//
#include <hip/hip_runtime.h>

#define N_FEATURES 1024
#define N_PROTOS   160
#define N_CLASSES  10
#define PPC        16
#define BATCH      8192

#define ZSTR 1026   // z-tile row stride (floats): 1026%64=2 -> conflict-free frag reads
#define BSTR 516    // omega-slab stride: 516*4 % 16 == 0 (b128 dst), 516%64=4
#define PSTR 36     // zp-slab stride:    36*4 % 16 == 0, distinct banks

typedef __attribute__((ext_vector_type(2))) float v2f;
typedef __attribute__((ext_vector_type(8))) float v8f;
typedef __attribute__((address_space(3))) float lds_f;

static __device__ __forceinline__ v8f wmma_f32(v2f a, v2f b, v8f c) {
  // D = A(16x4) * B(4x16) + C, full fp32 (matches the f32 reference numerics)
  return __builtin_amdgcn_wmma_f32_16x16x4_f32(false, a, false, b, (short)0, c,
                                               false, false);
}

// CDNA5 async direct-to-LDS copy: 16 bytes per lane, memory -> LDS, no VGPR
// round trip; tracked by ASYNCcnt.
static __device__ __forceinline__ void async_ld16(const float* g, float* l) {
  unsigned loff = (unsigned)(uintptr_t)(lds_f*)l;   // LDS byte offset (AS3)
  asm volatile("global_load_async_to_lds_b128 %0, %1, off"
               :: "v"(loff), "v"(g)
               : "memory");
}

static __device__ __forceinline__ void wait_async0() {
#if __has_builtin(__builtin_amdgcn_s_wait_asynccnt)
  __builtin_amdgcn_s_wait_asynccnt(0);
#else
  asm volatile("s_wait_asynccnt 0x0" ::: "memory");
#endif
}

// ---------------------------------------------------------------------------
// Kernel 1: zp = prototypes @ omega^T   (160 x 1024 -> d_ws)
// One wave per 16(M) x 64(N) tile; fragments fed straight from global b64
// loads (row-major layout == WMMA f32 A/B fragment layout).
// ---------------------------------------------------------------------------
__global__ __launch_bounds__(32)
void gmlvq_proj_protos(const float* __restrict__ prot,
                       const float* __restrict__ omega,
                       float* __restrict__ zp) {
  const int lane = threadIdx.x & 31;
  const int lm   = lane & 15;
  const int kh   = lane >> 4;              // lane half selects K pair
  const int m0   = blockIdx.x * 16;        // 0..144
  const int n0   = blockIdx.y * 64;        // 0..960

  const float* arow = prot + (size_t)(m0 + lm) * N_FEATURES + 2 * kh;
  v8f acc[4] = {};
  for (int k = 0; k < N_FEATURES; k += 4) {
    v2f a = *(const v2f*)(arow + k);
#pragma unroll
    for (int j = 0; j < 4; ++j) {
      v2f b = *(const v2f*)(omega + (size_t)(n0 + 16 * j + lm) * N_FEATURES + k + 2 * kh);
      acc[j] = wmma_f32(a, b, acc[j]);
    }
  }
#pragma unroll
  for (int j = 0; j < 4; ++j)
#pragma unroll
    for (int e = 0; e < 8; ++e)
      zp[(size_t)(m0 + e + 8 * kh) * N_FEATURES + n0 + 16 * j + lm] = acc[j][e];
}

// ---------------------------------------------------------------------------
// Kernel 2 (fused): per block = 64 batch rows (4 waves x 16-row tiles).
//  Phase 1: z_tile = x_tile @ omega^T  -> LDS (never touches HBM)
//           omega K-slabs async-staged into LDS, shared by all 4 waves.
//  Phase 2: dist = |z|^2 + |zp|^2 - 2 z.zp^T via WMMA from LDS; norms
//           accumulated off the same fragments; class-min + argmin in-wave.
// ---------------------------------------------------------------------------
__global__ __launch_bounds__(128)
void gmlvq_fused(const float* __restrict__ x,
                 const float* __restrict__ omega,
                 const float* __restrict__ zp,
                 float* __restrict__ dist,
                 float* __restrict__ pred) {
  extern __shared__ float smem[];
  float* zall  = smem;                     // 4 x [16][ZSTR] z tiles (per wave)
  float* stage = smem + 4 * 16 * ZSTR;     // shared staging: omega / zp slabs

  const int tid  = threadIdx.x;
  const int wave = tid >> 5;
  const int lane = tid & 31;
  const int lm   = lane & 15;
  const int kh   = lane >> 4;

  float* zw = zall + wave * (16 * ZSTR);
  const int mBase = blockIdx.x * 64 + wave * 16;
  const float* xrow = x + (size_t)(mBase + lm) * N_FEATURES + 2 * kh;

  // ---------------- Phase 1: project x tile ----------------
  for (int nt = 0; nt < 64; ++nt) {        // 64 N-tiles of 16 cols
    const int n0 = nt * 16;
    v8f c = {};
    for (int ks = 0; ks < 2; ++ks) {       // two 512-wide K slabs
      const int k0 = ks * 512;
      __syncthreads();                     // prior slab fully consumed
      // async-stage omega[n0..n0+15][k0..k0+511] (2048 x b128 / 128 thr)
#pragma unroll
      for (int i = 0; i < 16; ++i) {
        int flat4 = tid + 128 * i;
        int row   = flat4 >> 7;            // 128 float4 per row
        int c4    = (flat4 & 127) << 2;
        async_ld16(omega + (size_t)(n0 + row) * N_FEATURES + k0 + c4,
                   stage + row * BSTR + c4);
      }
      wait_async0();
      __syncthreads();
      const float* xk = xrow + k0;
#pragma unroll 8
      for (int k = 0; k < 512; k += 4) {
        v2f a = *(const v2f*)(xk + k);                             // global (L1-hot)
        v2f b = *(const v2f*)(stage + lm * BSTR + k + 2 * kh);     // LDS
        c = wmma_f32(a, b, c);
      }
    }
#pragma unroll
    for (int e = 0; e < 8; ++e)            // D layout: M = e + 8*half, N = lm
      zw[(e + 8 * kh) * ZSTR + n0 + lm] = c[e];
  }

  // ---------------- Phase 2: distances + argmin ----------------
  v8f   acc[10] = {};
  float sqp[10] = {};
  float sqx = 0.f;
  for (int ks = 0; ks < 32; ++ks) {        // 32-wide K slabs of zp
    const int k0 = ks * 32;
    __syncthreads();
    // async-stage zp[0..159][k0..k0+31] (1280 x b128 / 128 thr)
#pragma unroll
    for (int i = 0; i < 10; ++i) {
      int flat4 = tid + 128 * i;
      int row   = flat4 >> 3;              // 8 float4 per row
      int c4    = (flat4 & 7) << 2;
      async_ld16(zp + (size_t)row * N_FEATURES + k0 + c4,
                 stage + row * PSTR + c4);
    }
    wait_async0();
    __syncthreads();
#pragma unroll
    for (int k = 0; k < 32; k += 4) {
      v2f a = *(const v2f*)(zw + lm * ZSTR + k0 + k + 2 * kh);
      sqx = fmaf(a[0], a[0], sqx);
      sqx = fmaf(a[1], a[1], sqx);
#pragma unroll
      for (int j = 0; j < 10; ++j) {       // N-tile j == class j (PPC==16)
        v2f b = *(const v2f*)(stage + (16 * j + lm) * PSTR + k + 2 * kh);
        sqp[j] = fmaf(b[0], b[0], sqp[j]);
        sqp[j] = fmaf(b[1], b[1], sqp[j]);
        acc[j] = wmma_f32(a, b, acc[j]);
      }
    }
  }

  // combine the two K-halves held by lane L and lane L^16 (wave32 shuffles)
  float sqxT = sqx + __shfl_xor(sqx, 16, 32);

  float bestv[8];
  int   bestc[8];
#pragma unroll
  for (int e = 0; e < 8; ++e) { bestv[e] = 3.4e38f; bestc[e] = 0; }

#pragma unroll
  for (int j = 0; j < 10; ++j) {
    float sqpT = sqp[j] + __shfl_xor(sqp[j], 16, 32);
#pragma unroll
    for (int e = 0; e < 8; ++e) {
      int   mrow = e + 8 * kh;                         // row of this D element
      float sxm  = __shfl(sqxT, mrow, 32);             // |z_m|^2 broadcast
      float d    = fmaf(-2.f, acc[j][e], sxm + sqpT);  // full distance
      dist[(size_t)(mBase + mrow) * N_PROTOS + 16 * j + lm] = d;
      // class-min: min over the 16 prototypes of class j (one lane half)
      float cm = d;
      cm = fminf(cm, __shfl_xor(cm, 1, 32));
      cm = fminf(cm, __shfl_xor(cm, 2, 32));
      cm = fminf(cm, __shfl_xor(cm, 4, 32));
      cm = fminf(cm, __shfl_xor(cm, 8, 32));
      if (cm < bestv[e]) { bestv[e] = cm; bestc[e] = j; }
    }
  }
#pragma unroll
  for (int e = 0; e < 8; ++e)
    if (lm == 0)
      pred[mBase + e + 8 * kh] = (float)bestc[e];
}

// ---------------------------------------------------------------------------
extern "C" void kernel_launch(void* const* d_in, const int* in_sizes, int n_in,
                              void* d_out, int out_size, void* d_ws, size_t ws_size,
                              hipStream_t stream) {
  const float* x     = (const float*)d_in[0];   // [8192,1024]
  const float* prot  = (const float*)d_in[1];   // [160,1024]
  const float* omega = (const float*)d_in[2];   // [1024,1024]
  float* dist = (float*)d_out;                          // [8192,160]
  float* pred = dist + (size_t)BATCH * N_PROTOS;        // [8192] as float
  float* zp   = (float*)d_ws;                           // 160*1024 f32 = 640 KiB

  gmlvq_proj_protos<<<dim3(N_PROTOS / 16, N_FEATURES / 64), 32, 0, stream>>>(
      prot, omega, zp);

  const size_t lds_bytes = (size_t)(4 * 16 * ZSTR + 16 * BSTR) * sizeof(float); // 295680 B < 320 KB
  gmlvq_fused<<<dim3(BATCH / 64), 128, lds_bytes, stream>>>(x, omega, zp, dist, pred);
}